// HumanReceiver_27281632264217
// MI455X (gfx1250) — compile-verified
//
#include <hip/hip_runtime.h>
#include <hip/hip_bf16.h>
#include <math.h>

// ---------------- problem constants ----------------
#define N_NODES   20000
#define N_GRAPHS  20
#define NODES_PER 1000
#define N_EDGES   320000
#define N_REL     8
#define N_BASES   4
#define FEAT      128
#define HID       256
#define EMB       128
#define K1        (N_REL * FEAT + FEAT)   // 1152  (S1 cols + root cols)
#define N2        (N_REL * EMB + EMB)     // 1152  (xr2 cols + root cols)

typedef __attribute__((ext_vector_type(2))) float v2f;
typedef __attribute__((ext_vector_type(8))) float v8f;

static __device__ __forceinline__ v8f wmma4(v2f a, v2f b, v8f c) {
  // D = A(16x4,f32) * B(4x16,f32) + C(16x16,f32), full fp32 matrix pipe
  return __builtin_amdgcn_wmma_f32_16x16x4_f32(
      /*neg_a=*/false, a, /*neg_b=*/false, b,
      /*c_mod=*/(short)0, c, /*reuse_a=*/false, /*reuse_b=*/false);
}

// -------- fold W1 = [comp1 (x) basis1 ; root1]  -> (1152 x 256) row-major ----
__global__ void build_w1_k(const float* __restrict__ comp, const float* __restrict__ basis,
                           const float* __restrict__ root, float* __restrict__ W) {
  int idx = blockIdx.x * blockDim.x + threadIdx.x;
  if (idx >= K1 * HID) return;
  int k = idx / HID, o = idx % HID;
  float v;
  if (k < N_REL * FEAT) {
    int r = k >> 7, i = k & 127;
    v = 0.f;
    #pragma unroll
    for (int b = 0; b < N_BASES; ++b)
      v = fmaf(comp[r * N_BASES + b], basis[(b * FEAT + i) * HID + o], v);
  } else {
    v = root[(k - N_REL * FEAT) * HID + o];
  }
  W[idx] = v;
}

// -------- fold W2 = [comp2 (x) basis2 | root2] -> (256 x 1152) row-major ----
__global__ void build_w2_k(const float* __restrict__ comp, const float* __restrict__ basis,
                           const float* __restrict__ root, float* __restrict__ W) {
  int idx = blockIdx.x * blockDim.x + threadIdx.x;
  if (idx >= HID * N2) return;
  int k = idx / N2, c = idx % N2;
  float v;
  if (c < N_REL * EMB) {
    int r = c >> 7, o = c & 127;
    v = 0.f;
    #pragma unroll
    for (int b = 0; b < N_BASES; ++b)
      v = fmaf(comp[r * N_BASES + b], basis[(b * HID + k) * EMB + o], v);
  } else {
    v = root[k * EMB + (c - N_REL * EMB)];
  }
  W[idx] = v;
}

// -------- edge degree count per (dst, rel) ----------
__global__ void count_k(const int* __restrict__ dst, const int* __restrict__ et,
                        float* __restrict__ cnt) {
  int e = blockIdx.x * blockDim.x + threadIdx.x;
  if (e >= N_EDGES) return;
  atomicAdd(&cnt[dst[e] * N_REL + et[e]], 1.0f);
}

__global__ void norm_k(float* __restrict__ nrm) {
  int i = blockIdx.x * blockDim.x + threadIdx.x;
  if (i >= N_NODES * N_REL) return;
  nrm[i] = 1.0f / fmaxf(nrm[i], 1.0f);
}

// -------- layer1 pre-aggregation: S1[dst, et, f] += node_x[src, f] ----------
__global__ void agg1_k(const int* __restrict__ src, const int* __restrict__ dst,
                       const int* __restrict__ et, const float* __restrict__ X,
                       float* __restrict__ S1) {
  long long tid = (long long)blockIdx.x * blockDim.x + threadIdx.x;
  if (tid >= (long long)N_EDGES * (FEAT / 4)) return;
  int e = (int)(tid >> 5);          // FEAT/4 == 32 quads
  int f = ((int)tid & 31) * 4;
  int s = src[e], d = dst[e], r = et[e];
  const float4 xv = *(const float4*)(X + (size_t)s * FEAT + f);
  float* out = S1 + (size_t)d * (N_REL * FEAT) + r * FEAT + f;
  atomicAdd(out + 0, xv.x);
  atomicAdd(out + 1, xv.y);
  atomicAdd(out + 2, xv.z);
  atomicAdd(out + 3, xv.w);
}

// -------- scale S1 rows by norm ----------
__global__ void scale1_k(float* __restrict__ S1, const float* __restrict__ nrm) {
  long long idx = (long long)blockIdx.x * blockDim.x + threadIdx.x;
  if (idx >= (long long)N_NODES * N_REL * FEAT) return;
  int d = (int)(idx >> 10);             // /1024
  int r = ((int)idx & 1023) >> 7;       // /128
  S1[idx] *= nrm[d * N_REL + r];
}

// -------- GEMM1: h = relu([S1 | node_x] @ W1 + b1), M=20000 K=1152 N=256 ----
// one wave per block; wave computes a 32x32 tile with 4 fp32 WMMA accumulators
__global__ void gemm1_k(const float* __restrict__ S1, const float* __restrict__ X,
                        const float* __restrict__ W, const float* __restrict__ bias,
                        float* __restrict__ H) {
  const int m0 = blockIdx.x * 32;
  const int n0 = blockIdx.y * 32;
  const int l  = threadIdx.x;
  const int lh = l & 15;
  const int hi = l >> 4;
  v8f c00 = {}, c01 = {}, c10 = {}, c11 = {};

  // K segment 1: S1 (lda = 1024), W rows [0,1024)
  for (int k = 0; k < N_REL * FEAT; k += 4) {
    v2f a0 = *(const v2f*)(S1 + (size_t)(m0 + lh)      * (N_REL * FEAT) + k + 2 * hi);
    v2f a1 = *(const v2f*)(S1 + (size_t)(m0 + 16 + lh) * (N_REL * FEAT) + k + 2 * hi);
    const float* wr = W + (size_t)(k + 2 * hi) * HID + n0;
    v2f b0, b1;
    b0.x = wr[lh];        b0.y = wr[HID + lh];
    b1.x = wr[16 + lh];   b1.y = wr[HID + 16 + lh];
    c00 = wmma4(a0, b0, c00);  c01 = wmma4(a0, b1, c01);
    c10 = wmma4(a1, b0, c10);  c11 = wmma4(a1, b1, c11);
  }
  // K segment 2: node_x (lda = 128), W rows [1024,1152)
  for (int k = 0; k < FEAT; k += 4) {
    v2f a0 = *(const v2f*)(X + (size_t)(m0 + lh)      * FEAT + k + 2 * hi);
    v2f a1 = *(const v2f*)(X + (size_t)(m0 + 16 + lh) * FEAT + k + 2 * hi);
    const float* wr = W + (size_t)(N_REL * FEAT + k + 2 * hi) * HID + n0;
    v2f b0, b1;
    b0.x = wr[lh];        b0.y = wr[HID + lh];
    b1.x = wr[16 + lh];   b1.y = wr[HID + 16 + lh];
    c00 = wmma4(a0, b0, c00);  c01 = wmma4(a0, b1, c01);
    c10 = wmma4(a1, b0, c10);  c11 = wmma4(a1, b1, c11);
  }
  // epilogue: +bias, relu.  D layout: vgpr j -> m = j + 8*hi, n = lane&15
  const int ca = n0 + lh, cb = n0 + 16 + lh;
  const float ba = bias[ca], bb = bias[cb];
  #pragma unroll
  for (int j = 0; j < 8; ++j) {
    int ra = m0 + j + 8 * hi;
    int rb = ra + 16;
    H[(size_t)ra * HID + ca] = fmaxf(c00[j] + ba, 0.f);
    H[(size_t)ra * HID + cb] = fmaxf(c01[j] + bb, 0.f);
    H[(size_t)rb * HID + ca] = fmaxf(c10[j] + ba, 0.f);
    H[(size_t)rb * HID + cb] = fmaxf(c11[j] + bb, 0.f);
  }
}

// -------- GEMM2: [xr2 | root_part] = h @ W2, M=20000 K=256 N=1152 ----------
// cols [0,1024) -> XR (ld 1024); cols [1024,1152) -> NE = val + bias2
__global__ void gemm2_k(const float* __restrict__ H, const float* __restrict__ W,
                        const float* __restrict__ bias2, float* __restrict__ XR,
                        float* __restrict__ NE) {
  const int m0 = blockIdx.x * 32;
  const int n0 = blockIdx.y * 32;
  const int l  = threadIdx.x;
  const int lh = l & 15;
  const int hi = l >> 4;
  v8f c00 = {}, c01 = {}, c10 = {}, c11 = {};

  for (int k = 0; k < HID; k += 4) {
    v2f a0 = *(const v2f*)(H + (size_t)(m0 + lh)      * HID + k + 2 * hi);
    v2f a1 = *(const v2f*)(H + (size_t)(m0 + 16 + lh) * HID + k + 2 * hi);
    const float* wr = W + (size_t)(k + 2 * hi) * N2 + n0;
    v2f b0, b1;
    b0.x = wr[lh];        b0.y = wr[N2 + lh];
    b1.x = wr[16 + lh];   b1.y = wr[N2 + 16 + lh];
    c00 = wmma4(a0, b0, c00);  c01 = wmma4(a0, b1, c01);
    c10 = wmma4(a1, b0, c10);  c11 = wmma4(a1, b1, c11);
  }
  const int ca = n0 + lh, cb = n0 + 16 + lh;  // tile is uniformly xr2 or root
  if (n0 < N_REL * EMB) {
    #pragma unroll
    for (int j = 0; j < 8; ++j) {
      int ra = m0 + j + 8 * hi;
      int rb = ra + 16;
      XR[(size_t)ra * (N_REL * EMB) + ca] = c00[j];
      XR[(size_t)ra * (N_REL * EMB) + cb] = c01[j];
      XR[(size_t)rb * (N_REL * EMB) + ca] = c10[j];
      XR[(size_t)rb * (N_REL * EMB) + cb] = c11[j];
    }
  } else {
    const int oa = ca - N_REL * EMB, ob = cb - N_REL * EMB;
    const float ba = bias2[oa], bb = bias2[ob];
    #pragma unroll
    for (int j = 0; j < 8; ++j) {
      int ra = m0 + j + 8 * hi;
      int rb = ra + 16;
      NE[(size_t)ra * EMB + oa] = c00[j] + ba;
      NE[(size_t)ra * EMB + ob] = c01[j] + bb;
      NE[(size_t)rb * EMB + oa] = c10[j] + ba;
      NE[(size_t)rb * EMB + ob] = c11[j] + bb;
    }
  }
}

// -------- layer2 aggregation: NE[dst] += norm * xr2[src, et, :] -------------
__global__ void agg2_k(const int* __restrict__ src, const int* __restrict__ dst,
                       const int* __restrict__ et, const float* __restrict__ XR,
                       const float* __restrict__ nrm, float* __restrict__ NE) {
  long long tid = (long long)blockIdx.x * blockDim.x + threadIdx.x;
  if (tid >= (long long)N_EDGES * (EMB / 4)) return;
  int e = (int)(tid >> 5);
  int f = ((int)tid & 31) * 4;
  int s = src[e], d = dst[e], r = et[e];
  float nm = nrm[d * N_REL + r];
  const float4 v = *(const float4*)(XR + (size_t)s * (N_REL * EMB) + r * EMB + f);
  float* out = NE + (size_t)d * EMB + f;
  atomicAdd(out + 0, nm * v.x);
  atomicAdd(out + 1, nm * v.y);
  atomicAdd(out + 2, nm * v.z);
  atomicAdd(out + 3, nm * v.w);
}

// -------- message_emb = x @ msg_w + msg_b  (20 x 128) -----------------------
__global__ void msg_k(const float* __restrict__ x, const float* __restrict__ mw,
                      const float* __restrict__ mb, float* __restrict__ M) {
  int idx = blockIdx.x * blockDim.x + threadIdx.x;
  if (idx >= N_GRAPHS * EMB) return;
  int g = idx / EMB, o = idx % EMB;
  float s = mb[o];
  for (int k = 0; k < HID; ++k) s = fmaf(x[g * HID + k], mw[k * EMB + o], s);
  M[idx] = s;
}

// -------- scores[n] = dot(node_emb[n] - node_emb[nest[g]], msg[g]) ----------
__global__ void score_k(const float* __restrict__ NE, const float* __restrict__ M,
                        const int* __restrict__ batch, const int* __restrict__ nest,
                        float* __restrict__ SC) {
  int n = blockIdx.x * blockDim.x + threadIdx.x;
  if (n >= N_NODES) return;
  int g = batch[n];
  int ns = nest[g];
  const float* en = NE + (size_t)n * EMB;
  const float* ev = NE + (size_t)ns * EMB;
  const float* mg = M + (size_t)g * EMB;
  float s = 0.f;
  #pragma unroll 4
  for (int o = 0; o < EMB; ++o) s = fmaf(en[o] - ev[o], mg[o], s);
  SC[n] = s;
}

// -------- per-graph log_softmax over 1000 nodes -----------------------------
__global__ void lsm_k(const float* __restrict__ SC, float* __restrict__ OUT) {
  __shared__ float red[256];
  const int g = blockIdx.x, t = threadIdx.x;
  const float* s = SC + (size_t)g * NODES_PER;
  float mx = -INFINITY;
  for (int j = t; j < NODES_PER; j += 256) mx = fmaxf(mx, s[j]);
  red[t] = mx; __syncthreads();
  for (int w = 128; w > 0; w >>= 1) {
    if (t < w) red[t] = fmaxf(red[t], red[t + w]);
    __syncthreads();
  }
  mx = red[0]; __syncthreads();
  float se = 0.f;
  for (int j = t; j < NODES_PER; j += 256) se += expf(s[j] - mx);
  red[t] = se; __syncthreads();
  for (int w = 128; w > 0; w >>= 1) {
    if (t < w) red[t] += red[t + w];
    __syncthreads();
  }
  const float lse = logf(red[0]);
  for (int j = t; j < NODES_PER; j += 256)
    OUT[(size_t)g * NODES_PER + j] = s[j] - mx - lse;
}

// ---------------------------------------------------------------------------
extern "C" void kernel_launch(void* const* d_in, const int* in_sizes, int n_in,
                              void* d_out, int out_size, void* d_ws, size_t ws_size,
                              hipStream_t stream) {
  const float* x       = (const float*)d_in[0];
  // d_in[1] edge_attr: dead code in the reference (binning never used)
  const float* node_x  = (const float*)d_in[2];
  const int*   eidx    = (const int*)d_in[3];
  const int*   etype   = (const int*)d_in[4];
  const int*   batch   = (const int*)d_in[5];
  const int*   nest    = (const int*)d_in[6];
  const float* comp1   = (const float*)d_in[7];
  const float* basis1  = (const float*)d_in[8];
  const float* root1   = (const float*)d_in[9];
  const float* bias1   = (const float*)d_in[10];
  const float* comp2   = (const float*)d_in[11];
  const float* basis2  = (const float*)d_in[12];
  const float* root2   = (const float*)d_in[13];
  const float* bias2   = (const float*)d_in[14];
  const float* msg_w   = (const float*)d_in[15];
  const float* msg_b   = (const float*)d_in[16];
  float* out = (float*)d_out;

  const int* src = eidx;            // row 0
  const int* dst = eidx + N_EDGES;  // row 1

  // ---- workspace carving (floats) ----
  float* ws   = (float*)d_ws;
  float* BIG  = ws;                                   // 20000*1024  (S1, then xr2)
  float* H    = BIG  + (size_t)N_NODES * N_REL * FEAT; // 20000*256
  float* NE   = H    + (size_t)N_NODES * HID;          // 20000*128
  float* NRM  = NE   + (size_t)N_NODES * EMB;          // 160000
  float* W1   = NRM  + (size_t)N_NODES * N_REL;        // 1152*256
  float* W2   = W1   + (size_t)K1 * HID;               // 256*1152
  float* MSG  = W2   + (size_t)HID * N2;               // 20*128
  float* SC   = MSG  + (size_t)N_GRAPHS * EMB;         // 20000

  // ---- zero accumulation buffers ----
  hipMemsetAsync(BIG, 0, (size_t)N_NODES * N_REL * FEAT * sizeof(float), stream);
  hipMemsetAsync(NRM, 0, (size_t)N_NODES * N_REL * sizeof(float), stream);

  // ---- degree counts -> norm ----
  count_k<<<(N_EDGES + 255) / 256, 256, 0, stream>>>(dst, etype, NRM);
  norm_k<<<(N_NODES * N_REL + 255) / 256, 256, 0, stream>>>(NRM);

  // ---- weight folds ----
  build_w1_k<<<(K1 * HID + 255) / 256, 256, 0, stream>>>(comp1, basis1, root1, W1);
  build_w2_k<<<(HID * N2 + 255) / 256, 256, 0, stream>>>(comp2, basis2, root2, W2);

  // ---- layer 1: aggregate-first, then WMMA GEMM ----
  agg1_k<<<(N_EDGES * (FEAT / 4) + 255) / 256, 256, 0, stream>>>(src, dst, etype, node_x, BIG);
  scale1_k<<<((long long)N_NODES * N_REL * FEAT + 255) / 256, 256, 0, stream>>>(BIG, NRM);
  {
    dim3 grid(N_NODES / 32, HID / 32);   // 625 x 8
    gemm1_k<<<grid, 32, 0, stream>>>(BIG, node_x, W1, bias1, H);
  }

  // ---- layer 2: WMMA GEMM first (xr2 + root part), then aggregate ----
  {
    dim3 grid(N_NODES / 32, N2 / 32);    // 625 x 36
    gemm2_k<<<grid, 32, 0, stream>>>(H, W2, bias2, BIG /*xr2*/, NE);
  }
  agg2_k<<<(N_EDGES * (EMB / 4) + 255) / 256, 256, 0, stream>>>(src, dst, etype, BIG, NRM, NE);

  // ---- scoring + log_softmax ----
  msg_k<<<(N_GRAPHS * EMB + 255) / 256, 256, 0, stream>>>(x, msg_w, msg_b, MSG);
  score_k<<<(N_NODES + 255) / 256, 256, 0, stream>>>(NE, MSG, batch, nest, SC);
  lsm_k<<<N_GRAPHS, 256, 0, stream>>>(SC, out);
}